// DisLoss_70222715290003
// MI455X (gfx1250) — compile-verified
//
#include <hip/hip_runtime.h>

// DisLoss: out = (1/B) * sum_{b,k} attr[b,k] * || emb[idx[b,k]] - eb[b] ||^2
// B=4096, K=50, D=128, V=1e6. Memory-bound: 104.9 MB gather -> ~4.5us floor
// at 23.3 TB/s. Squared distances computed on the matrix pipe via
// V_WMMA_F32_16X16X4_F32: the f32 16x4 A-layout and 4x16 B-layout of the same
// 16 difference vectors coincide per-lane, so wmma(a, a, c) accumulates the
// Gram matrix diff*diff^T whose diagonal is the 16 squared L2 distances.

typedef __attribute__((ext_vector_type(2))) float v2f;
typedef __attribute__((ext_vector_type(8))) float v8f;

#define D_DIM 128
#define MAX_WAVES 8

__global__ __launch_bounds__(256) void disloss_main_kernel(
    const float* __restrict__ emb_batch,     // [B,128]
    const float* __restrict__ embedding,     // [V,128]
    const float* __restrict__ attr_sim,      // [B,K]
    const int*   __restrict__ indices,       // [B,K]
    float*       __restrict__ block_partials,// [B]
    int K)
{
    __shared__ __align__(16) float e_lds[D_DIM];          // this sample's embedding
    __shared__ __align__(16) float cmat[MAX_WAVES][32][8];// per-wave C dump
    __shared__ float wsum[MAX_WAVES];

    const int b      = blockIdx.x;
    const int tid    = threadIdx.x;
    const int wave   = tid >> 5;
    const int lane   = tid & 31;
    const int nwaves = blockDim.x >> 5;

    // Stage e_b into LDS (512 B, read K times from LDS instead of HBM).
    for (int i = tid; i < D_DIM; i += blockDim.x)
        e_lds[i] = emb_batch[(size_t)b * D_DIM + i];
    __syncthreads();

    // f32 16x4 A layout: lanes 0-15 hold row m = lane, d-pair {4s, 4s+1};
    // lanes 16-31 hold row m = lane-16, d-pair {4s+2, 4s+3}.
    const int m   = lane & 15;               // local neighbor row 0..15
    const int off = (lane >= 16) ? 2 : 0;    // d-pair offset within group of 4
    const int k   = wave * 16 + m;           // global neighbor index
    const int kc  = (k < K) ? k : (K - 1);   // clamp -> safe gather for padding
    const float attr = (k < K && lane < 16)
                         ? attr_sim[(size_t)b * K + k] : 0.0f; // mask padding
    const size_t row = (size_t)indices[(size_t)b * K + kc] * (size_t)D_DIM;
    const float* __restrict__ g = embedding + row;

    // Accumulate Gram(diff) over D in 32 steps of k=4. Each gathered row is
    // read exactly once (lane halves split even/odd d-pairs -> 512B per row).
    v8f c = {};
    #pragma unroll 4
    for (int s = 0; s < 32; ++s) {
        const int d0 = 4 * s + off;
        v2f gd = *(const v2f*)(g + d0);        // global_load_b64
        v2f ed = *(const v2f*)(&e_lds[d0]);    // ds_load_b64
        v2f a;
        a.x = gd.x - ed.x;
        a.y = gd.y - ed.y;
        // D = A * A^T + C  (A-frag == B-frag per-lane for this f32 shape)
        c = __builtin_amdgcn_wmma_f32_16x16x4_f32(
                false, a, false, a, (short)0, c, false, false);
    }

    // Dump C to LDS and pull the diagonal D[m][m] = ||diff_m||^2.
    // C layout: vgpr j: lanes 0-15 = (M=j, N=lane); lanes 16-31 = (M=j+8, N=lane-16).
    #pragma unroll
    for (int j = 0; j < 8; ++j)
        cmat[wave][lane][j] = c[j];
    __syncthreads();

    float val = 0.0f;
    if (lane < 16) {
        float sq = (m < 8) ? cmat[wave][m][m]          // (M=m, N=m), lane m
                           : cmat[wave][m + 16][m - 8];// (M=m, N=m), lane m+16
        val = sq * attr;
    }
    // Wave reduction over 32 lanes (upper half holds zeros).
    #pragma unroll
    for (int s = 16; s >= 1; s >>= 1)
        val += __shfl_xor(val, s, 32);
    if (lane == 0) wsum[wave] = val;
    __syncthreads();

    if (tid == 0) {
        float t = 0.0f;
        for (int w = 0; w < nwaves; ++w) t += wsum[w];
        block_partials[b] = t;
    }
}

// Deterministic final reduction (no float atomics; overwrites poisoned d_out).
__global__ __launch_bounds__(256) void disloss_reduce_kernel(
    const float* __restrict__ partials, float* __restrict__ out,
    int n, float inv_b)
{
    __shared__ float s[256];
    float t = 0.0f;
    for (int i = threadIdx.x; i < n; i += 256)
        t += partials[i];
    s[threadIdx.x] = t;
    __syncthreads();
    #pragma unroll
    for (int w = 128; w >= 1; w >>= 1) {
        if ((int)threadIdx.x < w) s[threadIdx.x] += s[threadIdx.x + w];
        __syncthreads();
    }
    if (threadIdx.x == 0) out[0] = s[0] * inv_b;
}

extern "C" void kernel_launch(void* const* d_in, const int* in_sizes, int n_in,
                              void* d_out, int out_size, void* d_ws, size_t ws_size,
                              hipStream_t stream) {
    const float* emb_batch = (const float*)d_in[0]; // [B,128] f32
    const float* embedding = (const float*)d_in[1]; // [V,128] f32
    const float* attr_sim  = (const float*)d_in[2]; // [B,K]   f32
    const int*   indices   = (const int*)d_in[3];   // [B,K]   i32
    // d_in[4] = beta (unused by the reference math)

    const int B = in_sizes[0] / D_DIM;
    const int K = in_sizes[2] / B;
    int kt = (K + 15) / 16;
    if (kt > MAX_WAVES) kt = MAX_WAVES;  // layout limit (K <= 128; here K = 50 -> 4 waves)

    float* partials = (float*)d_ws;      // B floats of scratch

    disloss_main_kernel<<<B, kt * 32, 0, stream>>>(
        emb_batch, embedding, attr_sim, indices, partials, K);
    disloss_reduce_kernel<<<1, 256, 0, stream>>>(
        partials, (float*)d_out, B, 1.0f / (float)B);
}